// MultiHeadGraphAttention_14130442404178
// MI455X (gfx1250) — compile-verified
//
#include <hip/hip_runtime.h>

// ---------------------------------------------------------------------------
// Multi-head graph attention for MI455X (gfx1250, wave32).
// GEMMs use V_WMMA_F32_16X16X4_F32 (full f32 precision on the matrix pipe).
// Edge phase uses float4 gathers + L2-resident f32 atomics (all intermediates
// fit in the 192MB L2, so gather/scatter stays on-chip).
// ---------------------------------------------------------------------------

typedef __attribute__((ext_vector_type(2))) float v2f;
typedef __attribute__((ext_vector_type(4))) float v4f;
typedef __attribute__((ext_vector_type(8))) float v8f;

#define NN  100000   // nodes (divisible by 16 -> no GEMM tail handling)
#define NE  1000000  // edges
#define DM  64       // model dim (= H * DH)
#define NH  4        // heads
#define DH  16       // head dim

// ---------------------------------------------------------------------------
// Zero scratch: ctx accumulator, segment-max keys, segment-sum z.
// ---------------------------------------------------------------------------
__global__ void init_ws_kernel(float* __restrict__ ctx,
                               unsigned* __restrict__ maxkey,
                               float* __restrict__ z) {
  int i = blockIdx.x * blockDim.x + threadIdx.x;
  if (i < NN * DM) ctx[i] = 0.0f;
  if (i < NN * NH) { maxkey[i] = 0u; z[i] = 0.0f; }
}

// ---------------------------------------------------------------------------
// out[row, n] = sum_k A[row, k] * W[k, n] + bias[n],   A: [nRows x 64] row-major
// Block = 128 threads = 4 waves; each wave owns one 16-wide column tile of a
// 16-row block.  K=64 -> 16 chained V_WMMA_F32_16X16X4_F32 per wave.
//
// Fragment layouts per CDNA5 ISA 7.12.2:
//   A 16x4 : v0 = {lanes0-15: K+0, lanes16-31: K+2}, v1 = {K+1, K+3}, M=lane&15
//   B 4x16 : v0 = {lanes0-15: row K+0, lanes16-31: row K+2}, v1 = {K+1, K+3},
//            N = lane&15
//   C 16x16: VGPR j = rows {j (lanes0-15), j+8 (lanes16-31)}, N = lane&15
// ---------------------------------------------------------------------------
__global__ __launch_bounds__(128)
void gemm64_wmma_kernel(const float* __restrict__ A,
                        const float* __restrict__ W,
                        const float* __restrict__ bias,
                        float* __restrict__ out) {
  const int lane = threadIdx.x & 31;
  const int wave = threadIdx.x >> 5;     // 0..3 -> column tile
  const int row0 = blockIdx.x * 16;
  const int n0   = wave * 16;
  const int m    = lane & 15;            // M index for A-frag, N index for B/C
  const int half = lane >> 4;            // 0 or 1
  const int kA   = half * 2;             // upper lanes hold K+2 / K+3

  const float* __restrict__ arow = A + (size_t)(row0 + m) * DM;

  v8f c = {};
#pragma unroll
  for (int k0 = 0; k0 < DM; k0 += 4) {
    v2f a, b;
    a.x = arow[k0 + kA];
    a.y = arow[k0 + kA + 1];
    b.x = W[(size_t)(k0 + kA) * DM + n0 + m];
    b.y = W[(size_t)(k0 + kA + 1) * DM + n0 + m];
    // (neg_a, A, neg_b, B, c_mod, C, reuse_a, reuse_b)
    c = __builtin_amdgcn_wmma_f32_16x16x4_f32(false, a, false, b,
                                              (short)0, c, false, false);
  }

  const float bb = bias[n0 + m];
#pragma unroll
  for (int j = 0; j < 8; ++j) {
    out[(size_t)(row0 + j + half * 8) * DM + n0 + m] = c[j] + bb;
  }
}

// Monotone float -> uint encoding so atomicMax(u32) orders like f32.
__device__ __forceinline__ unsigned enc_f32(float f) {
  int i = __float_as_int(f);
  return (unsigned)(i ^ ((i >> 31) | 0x80000000));
}
__device__ __forceinline__ float dec_f32(unsigned k) {
  return (k & 0x80000000u) ? __uint_as_float(k ^ 0x80000000u)
                           : __uint_as_float(~k);
}

// ---------------------------------------------------------------------------
// scores[e,h] = <qks[recv, h, :], qks[send, h, :]>;  atomic segment-max.
// One thread per (edge, head).  qks is L2-resident (25.6 MB).
// ---------------------------------------------------------------------------
__global__ void edge_scores_kernel(const float* __restrict__ qks,
                                   const int* __restrict__ senders,
                                   const int* __restrict__ receivers,
                                   float* __restrict__ attn_buf,
                                   unsigned* __restrict__ maxkey) {
  int idx = blockIdx.x * blockDim.x + threadIdx.x;
  if (idx >= NE * NH) return;
  const int e = idx >> 2, h = idx & 3;
  const int s = senders[e], r = receivers[e];
  const v4f* __restrict__ qr = (const v4f*)(qks + (size_t)r * DM + h * DH);
  const v4f* __restrict__ qs = (const v4f*)(qks + (size_t)s * DM + h * DH);
  float acc = 0.0f;
#pragma unroll
  for (int i = 0; i < 4; ++i) {
    v4f a = qr[i], b = qs[i];
    acc += a.x * b.x + a.y * b.y + a.z * b.z + a.w * b.w;
  }
  attn_buf[idx] = acc;
  atomicMax(&maxkey[r * NH + h], enc_f32(acc));
}

// ---------------------------------------------------------------------------
// e = exp(score - segmax);  atomic segment-sum z.
// ---------------------------------------------------------------------------
__global__ void edge_exp_kernel(const int* __restrict__ receivers,
                                const unsigned* __restrict__ maxkey,
                                float* __restrict__ attn_buf,
                                float* __restrict__ z) {
  int idx = blockIdx.x * blockDim.x + threadIdx.x;
  if (idx >= NE * NH) return;
  const int e = idx >> 2, h = idx & 3;
  const int r = receivers[e];
  const float mx = dec_f32(maxkey[r * NH + h]);
  const float ev = __expf(attn_buf[idx] - mx);
  attn_buf[idx] = ev;
  atomicAdd(&z[r * NH + h], ev);
}

// ---------------------------------------------------------------------------
// attn = e / z;  ctx[recv, h, :] += attn * vs[send, h, :]  (atomic scatter).
// ---------------------------------------------------------------------------
__global__ void edge_scatter_kernel(const float* __restrict__ vs,
                                    const int* __restrict__ senders,
                                    const int* __restrict__ receivers,
                                    const float* __restrict__ z,
                                    float* __restrict__ attn_buf,
                                    float* __restrict__ ctx) {
  int idx = blockIdx.x * blockDim.x + threadIdx.x;
  if (idx >= NE * NH) return;
  const int e = idx >> 2, h = idx & 3;
  const int r = receivers[e];
  const float a = attn_buf[idx] / z[r * NH + h];
  attn_buf[idx] = a;                       // final normalized attention
  const int s = senders[e];
  const v4f* __restrict__ vrow = (const v4f*)(vs + (size_t)s * DM + h * DH);
  float* __restrict__ crow = ctx + (size_t)r * DM + h * DH;
#pragma unroll
  for (int i = 0; i < 4; ++i) {
    v4f v = vrow[i];
    atomicAdd(&crow[i * 4 + 0], a * v.x);
    atomicAdd(&crow[i * 4 + 1], a * v.y);
    atomicAdd(&crow[i * 4 + 2], a * v.z);
    atomicAdd(&crow[i * 4 + 3], a * v.w);
  }
}

// ---------------------------------------------------------------------------
// inputs: nodes, W_qk, b_qk, W_v, b_v, W_out, b_out, senders, receivers
// d_out : [ out (NN*64 f32) | attn (NE*4 f32) ]
// d_ws  : [ qks | vs | ctx | maxkey(u32) | z ]  ~ 80.0 MB
// ---------------------------------------------------------------------------
extern "C" void kernel_launch(void* const* d_in, const int* in_sizes, int n_in,
                              void* d_out, int out_size, void* d_ws, size_t ws_size,
                              hipStream_t stream) {
  const float* nodes     = (const float*)d_in[0];
  const float* W_qk      = (const float*)d_in[1];
  const float* b_qk      = (const float*)d_in[2];
  const float* W_v       = (const float*)d_in[3];
  const float* b_v       = (const float*)d_in[4];
  const float* W_out     = (const float*)d_in[5];
  const float* b_out     = (const float*)d_in[6];
  const int*   senders   = (const int*)d_in[7];
  const int*   receivers = (const int*)d_in[8];

  float* out  = (float*)d_out;             // NN*DM
  float* attn = out + (size_t)NN * DM;     // NE*NH (also used as score scratch)

  float* ws      = (float*)d_ws;
  float* qks     = ws;
  float* vs      = ws + (size_t)NN * DM;
  float* ctx     = ws + (size_t)2 * NN * DM;
  unsigned* mkey = (unsigned*)(ws + (size_t)3 * NN * DM);
  float* z       = ws + (size_t)3 * NN * DM + (size_t)NN * NH;

  const int initGrid = (NN * DM + 255) / 256;
  const int gemmGrid = NN / 16;                 // 6250 row tiles
  const int edgeGrid = (NE * NH + 255) / 256;   // 15625 blocks

  init_ws_kernel<<<initGrid, 256, 0, stream>>>(ctx, mkey, z);

  gemm64_wmma_kernel<<<gemmGrid, 128, 0, stream>>>(nodes, W_qk, b_qk, qks);
  gemm64_wmma_kernel<<<gemmGrid, 128, 0, stream>>>(nodes, W_v,  b_v,  vs);

  edge_scores_kernel<<<edgeGrid, 256, 0, stream>>>(qks, senders, receivers, attn, mkey);
  edge_exp_kernel<<<edgeGrid, 256, 0, stream>>>(receivers, mkey, attn, z);
  edge_scatter_kernel<<<edgeGrid, 256, 0, stream>>>(vs, senders, receivers, z, attn, ctx);

  gemm64_wmma_kernel<<<gemmGrid, 128, 0, stream>>>(ctx, W_out, b_out, out);
}